// PreampGainLayer_89747636617523
// MI455X (gfx1250) — compile-verified
//
#include <hip/hip_runtime.h>
#include <math.h>

// ---------------------------------------------------------------------------
// MI455X (gfx1250) implementation of the DK-method preamp FFT filter.
//
//   out = irfft( rfft(concat(state[:,L:], x)) * H(p) )[:, -L:]
//
//   Kernel 1 (preamp_coef): per-row f64 circuit algebra -> 6 biquad coeffs.
//   Kernel 2 (preamp_fft):  per row-pair, pack 2 real rows into one complex
//     4096-pt FFT.  FFT = 3 radix-16 stages; each stage applies the DFT-16
//     matrix via chained V_WMMA_F32_16X16X4_F32.  All twiddles come from one
//     shared 4096-entry LDS table (quadrant-symmetric build).  Forward DIF ->
//     filter in digit-reversed order on conjugate slot pairs -> DIT inverse.
//   Data-buffer slots go through XOR bank swizzle SWZ() (conflict-free in all
//     six stage passes).  SWZ is hoisted to per-K-slice bases; pair elements
//     are derived via provably-equivalent +stride (stages 0/2, lets the
//     compiler re-merge DS accesses) or ^16 (stage 1).
// ---------------------------------------------------------------------------

typedef __attribute__((ext_vector_type(2))) float v2f;
typedef __attribute__((ext_vector_type(8))) float v8f;

#define TPB 256   // 8 waves of 32

__device__ __forceinline__ float2 cmulf(float2 a, float2 b) {
  return make_float2(a.x * b.x - a.y * b.y, a.x * b.y + a.y * b.x);
}

// Bank swizzle on complex-slot index (64 banks x 4B; slot = 8B = 2 banks).
// XORs bits {5,6,7}->1..3 and bit 9->4; preserves bit 0; bijective.
__device__ __forceinline__ int SWZ(int i) {
  return i ^ (((i >> 5) & 7) << 1) ^ (((i >> 9) & 1) << 4);
}

__device__ __forceinline__ v8f wmma_k4(v2f a, v2f b, v8f c) {
  // D = A(16x4 f32) * B(4x16 f32) + C(16x16 f32)
  return __builtin_amdgcn_wmma_f32_16x16x4_f32(
      /*neg_a=*/false, a, /*neg_b=*/false, b,
      /*c_mod=*/(short)0, c, /*reuse_a=*/false, /*reuse_b=*/false);
}

// One radix-16 stage over 256 columns of a 4096-element complex LDS buffer.
// Forward (dir=+1, DIF): load -> DFT16 matmul -> post-twiddle -> store.
// Inverse (dir=-1, DIT): load -> conj pre-twiddle -> conj-DFT16 -> store.
// Fim = Im(F_effective) slices, Fneg = -Im(F_effective).  wt[e]=e^{-2pi i e/4096}.
__device__ __forceinline__ void dft16_stage(float2* buf, const float2* wt,
                                            int stage, int dir,
                                            const v2f* Fr, const v2f* Fim,
                                            const v2f* Fneg,
                                            int wave, int lane) {
  const int half = lane >> 4;   // 0 or 1
  const int ln   = lane & 15;
  const int stride = (stage == 0) ? 256 : (stage == 1) ? 16 : 1;
  const int tmul   = (stage == 0) ? 1 : 16;   // exponent scale to base 4096
  const bool twid  = (stage != 2);

  for (int tt = 0; tt < 2; ++tt) {
    const int tile = wave + 8 * tt;      // 16 tiles of 16 columns per stage
    const int c = 16 * tile + ln;        // this lane's column for B/C tiles
    int colbase, j;
    if (stage == 0)      { colbase = c;                          j = c;      }
    else if (stage == 1) { colbase = ((c >> 4) << 8) + (c & 15); j = c & 15; }
    else                 { colbase = c << 4;                     j = 0;      }

    // Load B tile (complex), K-slice layout matching A: K = 4s + v + 2*half.
    // One SWZ per K-slice; v=1 element derived by +stride (st.0/2) or ^16 (st.1).
    v2f Br[4], Bi[4];
#pragma unroll
    for (int s = 0; s < 4; ++s) {
      const int b0 = SWZ(colbase + stride * (4 * s + 2 * half));
      const int b1 = (stage == 1) ? (b0 ^ 16) : (b0 + stride);
#pragma unroll
      for (int v = 0; v < 2; ++v) {
        const int k = 4 * s + v + 2 * half;
        float2 e = buf[v ? b1 : b0];
        if (dir < 0 && twid) {           // conj pre-twiddle e^{+2pi i jk/span}
          const float2 w = wt[tmul * j * k];
          e = cmulf(e, make_float2(w.x, -w.y));
        }
        Br[s][v] = e.x;
        Bi[s][v] = e.y;
      }
    }

    // Complex matmul: Cr = Fr*Br + (-Im)*Bi ; Ci = Fr*Bi + Im*Br
    v8f accR = {0.f, 0.f, 0.f, 0.f, 0.f, 0.f, 0.f, 0.f};
    v8f accI = {0.f, 0.f, 0.f, 0.f, 0.f, 0.f, 0.f, 0.f};
#pragma unroll
    for (int s = 0; s < 4; ++s) accR = wmma_k4(Fr[s],   Br[s], accR);
#pragma unroll
    for (int s = 0; s < 4; ++s) accR = wmma_k4(Fneg[s], Bi[s], accR);
#pragma unroll
    for (int s = 0; s < 4; ++s) accI = wmma_k4(Fr[s],   Bi[s], accI);
#pragma unroll
    for (int s = 0; s < 4; ++s) accI = wmma_k4(Fim[s],  Br[s], accI);

    // Post-twiddle (forward only) + scatter back in place, in row pairs.
#pragma unroll
    for (int g = 0; g < 4; ++g) {
      const int b0 = SWZ(colbase + stride * (2 * g + 8 * half));
      const int b1 = (stage == 1) ? (b0 ^ 16) : (b0 + stride);
#pragma unroll
      for (int d = 0; d < 2; ++d) {
        const int m = 2 * g + d;
        const int rr = m + 8 * half;     // C row M = m + 8*(lane/16)
        float2 o = make_float2(accR[m], accI[m]);
        if (dir > 0 && twid) {           // e^{-2pi i j*rr/span}
          o = cmulf(o, wt[tmul * j * rr]);
        }
        buf[d ? b1 : b0] = o;
      }
    }
  }
}

// H(z) = ((n0 z + n1) z + n2) / ((z + d1) z + d2)
__device__ __forceinline__ float2 crational(float2 z, float n0, float n1,
                                            float n2, float d1, float d2) {
  float2 num = cmulf(make_float2(n0 * z.x + n1, n0 * z.y), z);
  num.x += n2;
  float2 den = cmulf(make_float2(z.x + d1, z.y), z);
  den.x += d2;
  const float inv = 1.0f / (den.x * den.x + den.y * den.y);
  return make_float2((num.x * den.x + num.y * den.y) * inv,
                     (num.y * den.x - num.x * den.y) * inv);
}

__device__ __forceinline__ int rev16_3(int m) {  // base-16 digit reverse (3 digits)
  return ((m & 15) << 8) | (m & 240) | (m >> 8);
}

// ---------------------------------------------------------------------------
// Kernel 1: DK-method coefficient algebra (f64, one thread per batch row)
// ---------------------------------------------------------------------------
__global__ void preamp_coef_kernel(const float* __restrict__ cond,
                                   const float* __restrict__ a_rg,
                                   const float* __restrict__ a_r1,
                                   const float* __restrict__ a_c1,
                                   const float* __restrict__ a_c2,
                                   const float* __restrict__ cw,
                                   const float* __restrict__ cb,
                                   float* __restrict__ coef, int B) {
  const int r = blockIdx.x * blockDim.x + threadIdx.x;
  if (r >= B) return;

  const double T = 1.0 / 44100.0;
  const double RG = (0.9  + 0.2  / (1.0 + exp(-(double)a_rg[0]))) * 1.0e6;
  const double R1 = (0.99 + 0.02 / (1.0 + exp(-(double)a_r1[0]))) * 4.7e5;
  const double C1 = (0.9  + 0.2  / (1.0 + exp(-(double)a_c1[0]))) * 3.3e-9;
  const double C2 = (0.9  + 0.2  / (1.0 + exp(-(double)a_c2[0]))) * 1.0e-9;
  const double Gr = 1.0 / R1;
  const double g1 = 2.0 * C1 / T;
  const double g2 = 2.0 * C2 / T;

  double S[4][4] = {{ g1, -g1,      0.0, 1.0},
                    {-g1,  g1 + Gr, 0.0, 0.0},
                    {0.0,  0.0,     g2,  0.0},
                    {1.0,  0.0,     0.0, 0.0}};
  double a[4][8];
  for (int i = 0; i < 4; ++i)
    for (int jj = 0; jj < 4; ++jj) {
      a[i][jj] = S[i][jj];
      a[i][jj + 4] = (i == jj) ? 1.0 : 0.0;
    }
  for (int c = 0; c < 4; ++c) {
    const double pinv = 1.0 / a[c][c];
    for (int jj = 0; jj < 8; ++jj) a[c][jj] *= pinv;
    for (int rr = 0; rr < 4; ++rr)
      if (rr != c) {
        const double f = a[rr][c];
        for (int jj = 0; jj < 8; ++jj) a[rr][jj] -= f * a[c][jj];
      }
  }
  double Si[4][4];
  for (int i = 0; i < 4; ++i)
    for (int jj = 0; jj < 4; ++jj) Si[i][jj] = a[i][jj + 4];

  const double V[2][4]  = {{0, 1, -1, 0}, {0, 0, 1, 0}};   // Nvp rows
  const double Xr[2][4] = {{1, -1, 0, 0}, {0, 0, 1, 0}};   // Nxp rows

  double tv[2][4], tx[2][4], to[4];
  for (int aa = 0; aa < 2; ++aa)
    for (int jj = 0; jj < 4; ++jj) {
      double s1 = 0.0, s2 = 0.0;
      for (int i = 0; i < 4; ++i) {
        s1 += V[aa][i] * Si[i][jj];
        s2 += Xr[aa][i] * Si[i][jj];
      }
      tv[aa][jj] = s1;
      tx[aa][jj] = s2;
    }
  for (int jj = 0; jj < 4; ++jj) to[jj] = Si[2][jj];   // No = e3

  double Q[2][2], Ux[2][2], P[2][2], Uo[2], Uu[2], Do[2];
  for (int aa = 0; aa < 2; ++aa)
    for (int bb = 0; bb < 2; ++bb) {
      double q = 0.0, u = 0.0, pp = 0.0;
      for (int jj = 0; jj < 4; ++jj) {
        q  += tv[aa][jj] * V[bb][jj];
        u  += tx[aa][jj] * V[bb][jj];
        pp += tx[aa][jj] * Xr[bb][jj];
      }
      Q[aa][bb] = q; Ux[aa][bb] = u; P[aa][bb] = pp;
    }
  for (int bb = 0; bb < 2; ++bb) {
    double uo = 0.0, uu = 0.0, dd = 0.0;
    for (int jj = 0; jj < 4; ++jj) {
      uo += to[jj] * V[bb][jj];
      uu += Si[3][jj] * V[bb][jj];   // Nup' Si = row 3 of Si
      dd += to[jj] * Xr[bb][jj];
    }
    Uo[bb] = uo; Uu[bb] = uu; Do[bb] = dd;
  }

  const double gax[2] = {2.0 * g1, 2.0 * g2};
  double Ao[2][2], Bo[2], ZU[2][2];
  for (int aa = 0; aa < 2; ++aa) {
    for (int bb = 0; bb < 2; ++bb) {
      Ao[aa][bb] = gax[aa] * P[aa][bb] - ((aa == bb) ? 1.0 : 0.0);
      ZU[aa][bb] = gax[aa] * Ux[aa][bb];
    }
    Bo[aa] = gax[aa] * tx[aa][3];
  }
  const double Eo = to[3];

  const double t0 = (double)cond[r] * (double)cw[0] + (double)cb[0];
  const double pot = 1.0 / (1.0 + exp(-t0));
  double p = (pow(10.0, pot) - 1.0) / 9.0;
  p = fmin(fmax(p, 1.0e-4), 1.0 - 1.0e-4);

  const double M00 = (1.0 - p) * RG + Q[0][0], M01 = Q[0][1];
  const double M10 = Q[1][0],                  M11 = p * RG + Q[1][1];
  const double idet = 1.0 / (M00 * M11 - M01 * M10);
  const double RvQ[2][2] = {{ M11 * idet, -M01 * idet},
                            {-M10 * idet,  M00 * idet}};

  double W[2][2];
  for (int aa = 0; aa < 2; ++aa)
    for (int cc = 0; cc < 2; ++cc)
      W[aa][cc] = ZU[aa][0] * RvQ[0][cc] + ZU[aa][1] * RvQ[1][cc];

  double A2[2][2], Bm[2], UR[2], Dm[2];
  for (int aa = 0; aa < 2; ++aa) {
    for (int bb = 0; bb < 2; ++bb)
      A2[aa][bb] = Ao[aa][bb] - (W[aa][0] * Ux[bb][0] + W[aa][1] * Ux[bb][1]);
    Bm[aa] = Bo[aa] - (W[aa][0] * Uu[0] + W[aa][1] * Uu[1]);
  }
  for (int cc = 0; cc < 2; ++cc)
    UR[cc] = Uo[0] * RvQ[0][cc] + Uo[1] * RvQ[1][cc];
  for (int bb = 0; bb < 2; ++bb)
    Dm[bb] = Do[bb] - (UR[0] * Ux[bb][0] + UR[1] * Ux[bb][1]);
  const double Em = Eo - (UR[0] * Uu[0] + UR[1] * Uu[1]);

  const double trA  = A2[0][0] + A2[1][1];
  const double detA = A2[0][0] * A2[1][1] - A2[0][1] * A2[1][0];
  double Mx[2][2];
  for (int aa = 0; aa < 2; ++aa)
    for (int bb = 0; bb < 2; ++bb)
      Mx[aa][bb] = A2[aa][bb] - Bm[aa] * Dm[bb];
  const double trM  = Mx[0][0] + Mx[1][1];
  const double detM = Mx[0][0] * Mx[1][1] - Mx[0][1] * Mx[1][0];
  const double e1 = Em - 1.0;

  coef[r * 6 + 0] = (float)(1.0 + e1);               // num z^2
  coef[r * 6 + 1] = (float)(-trM + e1 * (-trA));     // num z^1
  coef[r * 6 + 2] = (float)(detM + e1 * detA);       // num z^0
  coef[r * 6 + 3] = 1.0f;                            // den z^2
  coef[r * 6 + 4] = (float)(-trA);                   // den z^1
  coef[r * 6 + 5] = (float)(detA);                   // den z^0
}

// ---------------------------------------------------------------------------
// Kernel 2: packed complex FFT(4096) -> rational filter -> IFFT per row-pair
// ---------------------------------------------------------------------------
__global__ __launch_bounds__(TPB) void preamp_fft_kernel(
    const float* __restrict__ x, const float* __restrict__ state,
    const float* __restrict__ coef, float* __restrict__ out) {
  __shared__ float2 zA[4096];   // data (in-place FFT / filter / IFFT), swizzled
  __shared__ float2 wt[4096];   // wt[e] = e^{-2*pi*i*e/4096}

  const int tid  = threadIdx.x;
  const int lane = tid & 31;
  const int wave = tid >> 5;
  const int r0 = 2 * blockIdx.x;
  const int r1 = r0 + 1;

  // ---- Twiddle table via quadrant symmetry: ~4 sincospif per thread --------
  for (int e = tid; e < 1024; e += TPB) {
    float sn, cs;
    sincospif((float)e * (1.0f / 2048.0f), &sn, &cs);
    wt[e]                 = make_float2( cs, -sn);   // w(e)
    wt[2048 - e]          = make_float2(-cs, -sn);   // w(2048-e) = -conj(w(e))
    wt[2048 + e]          = make_float2(-cs,  sn);   // w(2048+e) = -w(e)
    wt[(4096 - e) & 4095] = make_float2( cs,  sn);   // w(4096-e) = conj(w(e))
  }
  if (tid == 0) {
    wt[1024] = make_float2(0.f, -1.f);
    wt[3072] = make_float2(0.f,  1.f);
  }

  // ---- Stage input: z = st(row0) + i*st(row1), st = [state[2048:], x] ----
  {
    const float4* s0 = (const float4*)(state + (size_t)r0 * 4096 + 2048);
    const float4* s1 = (const float4*)(state + (size_t)r1 * 4096 + 2048);
    const float4* x0 = (const float4*)(x + (size_t)r0 * 2048);
    const float4* x1 = (const float4*)(x + (size_t)r1 * 2048);
    float4* zv = (float4*)zA;
    for (int i = tid; i < 512; i += TPB) {
      const float4 a0 = s0[i], a1 = s1[i];
      const int p0 = SWZ(4 * i);          // slots 4i,4i+1 stay adjacent
      zv[p0 >> 1]       = make_float4(a0.x, a1.x, a0.y, a1.y);
      zv[(p0 >> 1) ^ 1] = make_float4(a0.z, a1.z, a0.w, a1.w);  // SWZ(4i+2)=p0^2
      const float4 b0 = x0[i], b1 = x1[i];
      const int p1 = SWZ(2048 + 4 * i);
      zv[p1 >> 1]       = make_float4(b0.x, b1.x, b0.y, b1.y);
      zv[(p1 >> 1) ^ 1] = make_float4(b0.z, b1.z, b0.w, b1.w);
    }
  }
  __syncthreads();

  // ---- DFT-16 A-matrix K-slices from the table (period-16 exponent) ------
  // A layout: lane holds row M=lane&15; VGPR v & lane-half h -> K = 4s+v+2h.
  v2f Fr[4], Fi[4], Fn[4];
  {
    const int half = lane >> 4;
    const int M = lane & 15;
#pragma unroll
    for (int s = 0; s < 4; ++s)
#pragma unroll
      for (int v = 0; v < 2; ++v) {
        const int K = 4 * s + v + 2 * half;
        const float2 w = wt[((M * K) & 15) << 8];  // e^{-2pi i MK/16}
        Fr[s][v] = w.x;    // Re F
        Fi[s][v] = w.y;    // Im F (forward)
        Fn[s][v] = -w.y;   // -Im F (forward) == Im conj(F) (inverse)
      }
  }

  // ---- Forward DIF radix-16 (natural in -> digit-reversed out) -----------
  dft16_stage(zA, wt, 0, +1, Fr, Fi, Fn, wave, lane); __syncthreads();
  dft16_stage(zA, wt, 1, +1, Fr, Fi, Fn, wave, lane); __syncthreads();
  dft16_stage(zA, wt, 2, +1, Fr, Fi, Fn, wave, lane); __syncthreads();

  // ---- In-place rational filter on conjugate slot pairs -------------------
  const float n00 = coef[6 * r0 + 0], n01 = coef[6 * r0 + 1],
              n02 = coef[6 * r0 + 2], d01 = coef[6 * r0 + 4],
              d02 = coef[6 * r0 + 5];
  const float n10 = coef[6 * r1 + 0], n11 = coef[6 * r1 + 1],
              n12 = coef[6 * r1 + 2], d11 = coef[6 * r1 + 4],
              d12 = coef[6 * r1 + 5];
  const float sc = 0.5f / 4096.0f;   // unpack 1/2 and IFFT 1/N folded here
  for (int m = tid; m < 4096; m += TPB) {
    const int k = rev16_3(m);
    const int kneg = (4096 - k) & 4095;
    const int mh = rev16_3(kneg);
    if (mh < m) continue;            // pair (m, mh) owned by smaller slot
    const float2 z = wt[k];          // e^{-2pi i k/4096}
    const float2 H0 = crational(z, n00, n01, n02, d01, d02);
    const float2 H1 = crational(z, n10, n11, n12, d11, d12);
    const float2 Am = make_float2((H0.x + H1.x) * sc, (H0.y + H1.y) * sc);
    const float2 Bm = make_float2((H0.x - H1.x) * sc, (H0.y - H1.y) * sc);
    const int pm  = SWZ(m);
    const int pmh = SWZ(mh);
    const float2 Zm = zA[pm];
    const float2 Zh = zA[pmh];
    // Y[m] = Am*Z[m] + Bm*conj(Z[mh])
    const float2 cZh = make_float2(Zh.x, -Zh.y);
    const float2 ya = cmulf(Am, Zm);
    const float2 yb = cmulf(Bm, cZh);
    zA[pm] = make_float2(ya.x + yb.x, ya.y + yb.y);
    if (mh != m) {
      // Y[mh] = conj(Am)*Z[mh] + conj(Bm)*conj(Z[m])  (H(-k) = conj(H(k)))
      const float2 cAm = make_float2(Am.x, -Am.y);
      const float2 cBm = make_float2(Bm.x, -Bm.y);
      const float2 cZm = make_float2(Zm.x, -Zm.y);
      const float2 za2 = cmulf(cAm, Zh);
      const float2 zb2 = cmulf(cBm, cZm);
      zA[pmh] = make_float2(za2.x + zb2.x, za2.y + zb2.y);
    }
  }
  __syncthreads();

  // ---- Inverse DIT radix-16 (digit-reversed in -> natural out) -----------
  dft16_stage(zA, wt, 2, -1, Fr, Fn, Fi, wave, lane); __syncthreads();
  dft16_stage(zA, wt, 1, -1, Fr, Fn, Fi, wave, lane); __syncthreads();
  dft16_stage(zA, wt, 0, -1, Fr, Fn, Fi, wave, lane); __syncthreads();

  // ---- Keep last L=2048 samples; Re -> row0, Im -> row1 ------------------
  {
    float4* o0 = (float4*)(out + (size_t)r0 * 2048);
    float4* o1 = (float4*)(out + (size_t)r1 * 2048);
    const float4* zv = (const float4*)zA;
    for (int i = tid; i < 512; i += TPB) {
      const int p0 = SWZ(2048 + 4 * i);
      const float4 p = zv[p0 >> 1];
      const float4 q = zv[(p0 >> 1) ^ 1];
      o0[i] = make_float4(p.x, p.z, q.x, q.z);
      o1[i] = make_float4(p.y, p.w, q.y, q.w);
    }
  }
}

// ---------------------------------------------------------------------------
extern "C" void kernel_launch(void* const* d_in, const int* in_sizes, int n_in,
                              void* d_out, int out_size, void* d_ws,
                              size_t ws_size, hipStream_t stream) {
  const float* x     = (const float*)d_in[0];
  const float* cond  = (const float*)d_in[1];
  const float* state = (const float*)d_in[2];
  const float* a_rg  = (const float*)d_in[3];
  const float* a_r1  = (const float*)d_in[4];
  const float* a_c1  = (const float*)d_in[5];
  const float* a_c2  = (const float*)d_in[6];
  const float* cw    = (const float*)d_in[7];
  const float* cb    = (const float*)d_in[8];
  float* out  = (float*)d_out;
  float* coef = (float*)d_ws;          // B*6 floats of scratch

  const int B = in_sizes[1];           // cond has B elements

  preamp_coef_kernel<<<(B + 255) / 256, 256, 0, stream>>>(
      cond, a_rg, a_r1, a_c1, a_c2, cw, cb, coef, B);
  preamp_fft_kernel<<<B / 2, TPB, 0, stream>>>(x, state, coef, out);
}